// GaussianEMDKernel_85409719648942
// MI455X (gfx1250) — compile-verified
//
#include <hip/hip_runtime.h>
#include <math.h>

// GaussianEMD / sliced-Wasserstein, fused for MI455X (gfx1250):
//   per block: 16-projection GEMM tile via V_WMMA_F32_16X16X4_F32 into LDS,
//   in-LDS bitonic sort of 32 columns of 2048, deterministic reduction.
// Round 4: the scheduler was sinking A-fragment loads to ~2 ahead of their
// consuming WMMA (s_wait_loadcnt 0x0 everywhere). Fence the load batch with
// __builtin_amdgcn_sched_barrier(0) so all 32 loads issue first and the
// WMMA chain drains them with staggered waits.

typedef __attribute__((ext_vector_type(2))) float v2f;
typedef __attribute__((ext_vector_type(8))) float v8f;

#define NPTS     2048
#define DIM      64
#define PTILE    16              // projections per block (one WMMA N tile)
#define THREADS  256
#define BUF_ELEMS (NPTS * PTILE) // 32768 floats = 128 KB
#define BUF_PAD   16             // bank-shift second buffer

extern "C" __global__ __launch_bounds__(THREADS)
__attribute__((amdgpu_waves_per_eu(2)))
void swd_tile_kernel(const float* __restrict__ x,
                     const float* __restrict__ y,
                     const float* __restrict__ proj,
                     float* __restrict__ partials)
{
    extern __shared__ float sh[];                 // [BUF_ELEMS] + pad + [BUF_ELEMS]
    float* s1 = sh;
    float* s2 = sh + BUF_ELEMS + BUF_PAD;

    const int tid  = threadIdx.x;
    const int wave = tid >> 5;
    const int lane = tid & 31;
    const int blk  = blockIdx.x;                  // 0..2047
    const int b    = blk >> 9;                    // 512 tiles per batch
    const int p0   = (blk & 511) * PTILE;

    // ---- Load B fragments once: proj tile, 16 K-steps of 4 ----
    // B (KxN=4x16) layout mirrors A: lane L holds col N=L&15, K=(L>>4)*2+{0,1}
    v2f bf[16];
    {
        const float* prow = proj + (size_t)(p0 + (lane & 15)) * DIM + ((lane >> 4) << 1);
#pragma unroll
        for (int ks = 0; ks < 16; ++ks)
            bf[ks] = *(const v2f*)(prow + ks * 4);
    }

    const float* xb = x + (size_t)b * NPTS * DIM;
    const float* yb = y + (size_t)b * NPTS * DIM;

    // ---- GEMM phase: p1 = x[b]*projT -> s1, p2 = y[b]*projT -> s2 ----
    // Per wave: 16 M-tiles of 16 rows; each iteration computes the x-tile and
    // y-tile together as two independent WMMA accumulation chains.
#pragma unroll 1
    for (int t = 0; t < 16; ++t) {
        const int m0 = wave * 256 + t * 16;
        const size_t roff = (size_t)(m0 + (lane & 15)) * DIM + ((lane >> 4) << 1);
        const float* ax = xb + roff;
        const float* ay = yb + roff;
        __builtin_prefetch(ax + 16 * DIM, 0, 3);  // next M-tile, keep near
        __builtin_prefetch(ay + 16 * DIM, 0, 3);

        // Batch-issue all 32 A-fragment loads into distinct registers.
        v2f a0[16], a1[16];
#pragma unroll
        for (int ks = 0; ks < 16; ++ks) {
            a0[ks] = *(const v2f*)(ax + ks * 4);
            a1[ks] = *(const v2f*)(ay + ks * 4);
        }

        // Keep the scheduler from sinking loads into the WMMA chain:
        // everything above issues before anything below.
        __builtin_amdgcn_sched_barrier(0);

        v8f accx = {};
        v8f accy = {};
#pragma unroll
        for (int ks = 0; ks < 16; ++ks) {
            // (neg_a, A, neg_b, B, c_mod, C, reuse_a, reuse_b)
            accx = __builtin_amdgcn_wmma_f32_16x16x4_f32(
                     false, a0[ks], false, bf[ks], (short)0, accx, false, false);
            accy = __builtin_amdgcn_wmma_f32_16x16x4_f32(
                     false, a1[ks], false, bf[ks], (short)0, accy, false, false);
        }

        // D layout: VGPR r -> row m0 + r + (lane>=16 ? 8 : 0), col = lane&15
        const int col   = lane & 15;
        const int rbase = m0 + ((lane >> 4) << 3);
#pragma unroll
        for (int r = 0; r < 8; ++r) {
            s1[(rbase + r) * PTILE + col] = accx[r];
            s2[(rbase + r) * PTILE + col] = accy[r];
        }
    }
    __syncthreads();

    // ---- Bitonic sort: 32 columns (16 per buffer) of 2048 elements ----
    for (int k = 2; k <= NPTS; k <<= 1) {
        for (int j = k >> 1; j > 0; j >>= 1) {
            // 32 cols * 1024 pairs = 32768 pairs per stage
            for (int q = tid; q < 32 * (NPTS / 2); q += THREADS) {
                const int pairIdx = q >> 5;       // 0..1023
                const int cc      = q & 31;       // column across both buffers
                float* bufp = (cc & 16) ? s2 : s1;
                const int c = cc & 15;
                const int i = ((pairIdx & ~(j - 1)) << 1) | (pairIdx & (j - 1));
                const int l = i | j;
                const bool asc = ((i & k) == 0);
                float vi = bufp[i * PTILE + c];
                float vl = bufp[l * PTILE + c];
                if ((vi > vl) == asc) {
                    bufp[i * PTILE + c] = vl;
                    bufp[l * PTILE + c] = vi;
                }
            }
            __syncthreads();
        }
    }

    // ---- Accumulate sum of (p1_sorted - p2_sorted)^2 over this tile ----
    float local = 0.0f;
    for (int f = tid; f < BUF_ELEMS; f += THREADS) {
        float d = s1[f] - s2[f];
        local = fmaf(d, d, local);
    }

    __shared__ float red[THREADS];
    red[tid] = local;
    __syncthreads();
    for (int s = THREADS >> 1; s > 0; s >>= 1) {
        if (tid < s) red[tid] += red[tid + s];
        __syncthreads();
    }
    if (tid == 0) partials[blk] = red[0];
}

// One block per batch: deterministic fixed-tree reduction of 512 partials.
extern "C" __global__ __launch_bounds__(256)
void swd_finalize_kernel(const float* __restrict__ partials,
                         float* __restrict__ out)
{
    __shared__ float red[256];
    const int b   = blockIdx.x;
    const int tid = threadIdx.x;
    red[tid] = partials[b * 512 + tid] + partials[b * 512 + 256 + tid];
    __syncthreads();
    for (int s = 128; s > 0; s >>= 1) {
        if (tid < s) red[tid] += red[tid + s];
        __syncthreads();
    }
    if (tid == 0) {
        float swd = red[0] * (1.0f / (2048.0f * 8192.0f));  // mean over N*P
        out[b] = expf(-0.5f * swd * swd);                   // sigma = 1
    }
}

extern "C" void kernel_launch(void* const* d_in, const int* in_sizes, int n_in,
                              void* d_out, int out_size, void* d_ws, size_t ws_size,
                              hipStream_t stream) {
    (void)in_sizes; (void)n_in; (void)out_size; (void)ws_size;
    const float* x    = (const float*)d_in[0];   // [4,2048,64]
    const float* y    = (const float*)d_in[1];   // [4,2048,64]
    const float* proj = (const float*)d_in[2];   // [8192,64]
    float* out      = (float*)d_out;             // [4]
    float* partials = (float*)d_ws;              // 2048 floats (8 KB)

    const size_t shbytes = (size_t)(2 * BUF_ELEMS + BUF_PAD) * sizeof(float); // 256 KB + pad
    // gfx1250 WGP has 320 KB LDS; opt in to >default dynamic LDS (host call, not captured).
    (void)hipFuncSetAttribute((const void*)swd_tile_kernel,
                              hipFuncAttributeMaxDynamicSharedMemorySize, (int)shbytes);

    swd_tile_kernel<<<dim3(2048), dim3(THREADS), shbytes, stream>>>(x, y, proj, partials);
    swd_finalize_kernel<<<dim3(4), dim3(256), 0, stream>>>(partials, out);
}